// GraphLearningLayer_20177756357164
// MI455X (gfx1250) — compile-verified
//
#include <hip/hip_runtime.h>
#include <hip/hip_bf16.h>
#include <math.h>

typedef __attribute__((ext_vector_type(2))) float v2f;
typedef __attribute__((ext_vector_type(8))) float v8f;

// Problem constants (from reference setup)
#define BB 4
#define NN 512
#define DD 256
#define LL 64
#define TI 8      // i-rows per block in distance kernel
#define BLK 512   // threads per block in distance kernel (one j per thread)

// -----------------------------------------------------------------------------
// Kernel 1: xsT[b][l][n] = learn_w[l] * sum_d x[b][n][d] * W[l][d]
// One wave (32 threads) computes one 16x16 (n x l) tile via V_WMMA_F32_16X16X4_F32.
// A tile: x rows (M=n, K=d). B tile: B[k][l] = W[l][k].
// A-layout (f32 16x4): lane 0-15 -> M=lane, VGPR0=K0,VGPR1=K1; lanes 16-31 -> K2,K3.
// C-layout: VGPR r: lanes 0-15 -> M=r, N=lane; lanes 16-31 -> M=r+8, N=lane-16.
// -----------------------------------------------------------------------------
__global__ __launch_bounds__(32)
void proj_wmma_kernel(const float* __restrict__ x,
                      const float* __restrict__ W,
                      const float* __restrict__ lw,
                      float* __restrict__ xsT)
{
    const int b  = blockIdx.z;
    const int m0 = blockIdx.y * 16;   // n-tile base
    const int l0 = blockIdx.x * 16;   // l-tile base
    const int lane = threadIdx.x;     // 0..31 (wave32)
    const int ml = lane & 15;
    const int kh = (lane >> 4) * 2;   // 0 or 2: which K-pair this half-wave holds

    const float* xrow = x + ((size_t)b * NN + (m0 + ml)) * DD;  // A: row m0+ml
    const float* wrow = W + (size_t)(l0 + ml) * DD;             // B: column l0+ml (W is [L][D])

    v8f acc = {};
    #pragma unroll 4
    for (int k = 0; k < DD; k += 4) {
        v2f a = *(const v2f*)(xrow + k + kh);   // x[n][k+kh], x[n][k+kh+1]
        v2f bm = *(const v2f*)(wrow + k + kh);  // W[l][k+kh], W[l][k+kh+1]
        acc = __builtin_amdgcn_wmma_f32_16x16x4_f32(
            /*neg_a=*/false, a, /*neg_b=*/false, bm,
            /*c_mod=*/(short)0, acc, /*reuse_a=*/false, /*reuse_b=*/false);
    }

    // Fold nonnegative learn_w into the projection: xs = w[l] * x_hat
    const float scale = lw[l0 + ml];           // this lane's column l
    const int   nbase = m0 + (lane >> 4) * 8;  // C layout: M = r + 8*(lane>=16)
    float* outp = xsT + ((size_t)b * LL + (l0 + ml)) * NN + nbase;
    #pragma unroll
    for (int r = 0; r < 8; ++r)
        outp[r] = acc[r] * scale;
}

// -----------------------------------------------------------------------------
// Kernel 2: for a block of TI rows i, all j:
//   d  = leakyrelu( sum_l |xs[i,l]-xs[j,l]| )
//   e  = adj * exp(d - rowmax(d));  out = e / rowsum(e) + 1e-10
// -----------------------------------------------------------------------------
__device__ __forceinline__ float wave_max(float v) {
    #pragma unroll
    for (int off = 16; off > 0; off >>= 1)
        v = fmaxf(v, __shfl_xor(v, off, 32));
    return v;
}
__device__ __forceinline__ float wave_sum(float v) {
    #pragma unroll
    for (int off = 16; off > 0; off >>= 1)
        v += __shfl_xor(v, off, 32);
    return v;
}

__global__ __launch_bounds__(BLK)
void dist_softmax_kernel(const float* __restrict__ xsT,
                         const float* __restrict__ adj,
                         float* __restrict__ out)
{
    const int b  = blockIdx.x;
    const int i0 = blockIdx.y * TI;
    const int t  = threadIdx.x;       // 0..511, t == j
    const int j  = t;
    const int lane = t & 31;
    const int wv   = t >> 5;          // 0..15 waves

    __shared__ float rows[TI][LL];
    __shared__ float part[TI][16];

    // Stage the TI i-rows into LDS (512 elements, one per thread)
    {
        const int r = t >> 6;         // 0..7
        const int l = t & 63;         // 0..63
        rows[r][l] = xsT[((size_t)b * LL + l) * NN + (i0 + r)];
    }
    __syncthreads();

    // Load this thread's j-column (coalesced across lanes: stride-1 in n)
    float col[LL];
    #pragma unroll
    for (int l = 0; l < LL; ++l)
        col[l] = xsT[((size_t)b * LL + l) * NN + j];

    // Weighted L1 distances against each of the TI rows (abs folded in sub/add)
    float d[TI];
    #pragma unroll
    for (int r = 0; r < TI; ++r) {
        float s = 0.0f;
        #pragma unroll
        for (int l = 0; l < LL; ++l)
            s += fabsf(col[l] - rows[r][l]);
        d[r] = fmaxf(s, 0.01f * s);   // leaky_relu (s>=0 in practice, kept faithful)
    }

    // Row max over j (block-wide), per i-row
    #pragma unroll
    for (int r = 0; r < TI; ++r) {
        float v = wave_max(d[r]);
        if (lane == 0) part[r][wv] = v;
    }
    __syncthreads();
    float m[TI];
    #pragma unroll
    for (int r = 0; r < TI; ++r) {
        float v = part[r][0];
        #pragma unroll
        for (int k = 1; k < 16; ++k) v = fmaxf(v, part[r][k]);  // LDS broadcast
        m[r] = v;
    }
    __syncthreads();

    // e = adj * exp(d - max); row sum over j
    float e[TI];
    #pragma unroll
    for (int r = 0; r < TI; ++r) {
        const float a = adj[((size_t)b * NN + (i0 + r)) * NN + j];
        e[r] = a * __expf(d[r] - m[r]);
        float v = wave_sum(e[r]);
        if (lane == 0) part[r][wv] = v;
    }
    __syncthreads();
    #pragma unroll
    for (int r = 0; r < TI; ++r) {
        float s = part[r][0];
        #pragma unroll
        for (int k = 1; k < 16; ++k) s += part[r][k];
        // faithful to reference: epsilon added AFTER the division
        out[((size_t)b * NN + (i0 + r)) * NN + j] = e[r] / s + 1e-10f;
    }
}

// -----------------------------------------------------------------------------
// Launch
// -----------------------------------------------------------------------------
extern "C" void kernel_launch(void* const* d_in, const int* in_sizes, int n_in,
                              void* d_out, int out_size, void* d_ws, size_t ws_size,
                              hipStream_t stream)
{
    const float* x   = (const float*)d_in[0];   // [B,N,D]
    const float* adj = (const float*)d_in[1];   // [B,N,N]
    const float* W   = (const float*)d_in[2];   // [L,D]
    const float* lw  = (const float*)d_in[3];   // [L]
    float* out = (float*)d_out;                 // [B,N,N]
    float* xsT = (float*)d_ws;                  // [B,L,N] = 512 KB scratch

    // Kernel 1: one wave per 16x16 (n,l) tile
    dim3 g1(LL / 16, NN / 16, BB);   // (4, 32, 4)
    proj_wmma_kernel<<<g1, 32, 0, stream>>>(x, W, lw, xsT);

    // Kernel 2: one block per (b, 8 i-rows)
    dim3 g2(BB, NN / TI);            // (4, 64)
    dist_softmax_kernel<<<g2, BLK, 0, stream>>>(xsT, adj, out);
}